// ModifiedSimpleNetwork_33732673143508
// MI455X (gfx1250) — compile-verified
//
#include <hip/hip_runtime.h>
#include <math.h>

#define NNODES 10000
#define NEDGES 160000
#define NCGT 15

typedef __attribute__((ext_vector_type(2))) float v2f;
typedef __attribute__((ext_vector_type(8))) float v8f;

// ---------------------------------------------------------------------------
// CG table: flat float buffer, offsets for each (l1,l2,l3) triangle combo
// sizes: 1,9,25,9,9,27,45,45,75,25,45,75,25,75,125 -> total 615 floats
// ---------------------------------------------------------------------------
__host__ __device__ static inline int cg_offset(int l1, int l2, int l3) {
  const int T[NCGT][4] = {
      {0,0,0,0},  {0,1,1,1},  {0,2,2,10}, {1,0,1,35}, {1,1,0,44},
      {1,1,1,53}, {1,1,2,80}, {1,2,1,125},{1,2,2,170},{2,0,2,245},
      {2,1,1,270},{2,1,2,315},{2,2,0,390},{2,2,1,415},{2,2,2,490}};
  for (int i = 0; i < NCGT; ++i)
    if (T[i][0] == l1 && T[i][1] == l2 && T[i][2] == l3) return T[i][3];
  return 0;
}

// ---------------------------------------------------------------------------
// Device computation of real Clebsch-Gordan tensors (port of _real_cg)
// ---------------------------------------------------------------------------
__device__ static double dfact(int n) {
  double r = 1.0;
  for (int i = 2; i <= n; ++i) r *= (double)i;
  return r;
}

__device__ static double cgc(int j1, int m1, int j2, int m2, int j3, int m3) {
  if (m1 + m2 != m3) return 0.0;
  double pref = sqrt((double)(2 * j3 + 1) * dfact(j1 + j2 - j3) * dfact(j1 - j2 + j3) *
                     dfact(-j1 + j2 + j3) / dfact(j1 + j2 + j3 + 1));
  pref *= sqrt(dfact(j1 + m1) * dfact(j1 - m1) * dfact(j2 + m2) * dfact(j2 - m2) *
               dfact(j3 + m3) * dfact(j3 - m3));
  int kmin = 0;
  if (j2 - j3 - m1 > kmin) kmin = j2 - j3 - m1;
  if (j1 - j3 + m2 > kmin) kmin = j1 - j3 + m2;
  int kmax = j1 + j2 - j3;
  if (j1 - m1 < kmax) kmax = j1 - m1;
  if (j2 + m2 < kmax) kmax = j2 + m2;
  double s = 0.0;
  for (int k = kmin; k <= kmax; ++k) {
    double d = dfact(k) * dfact(j1 + j2 - j3 - k) * dfact(j1 - m1 - k) *
               dfact(j2 + m2 - k) * dfact(j3 - j2 + m1 + k) * dfact(j3 - j1 - m2 + k);
    s += ((k & 1) ? -1.0 : 1.0) / d;
  }
  return pref * s;
}

__device__ static void qmat(int l, double qr[5][5], double qi[5][5]) {
  for (int i = 0; i < 5; ++i)
    for (int j = 0; j < 5; ++j) { qr[i][j] = 0.0; qi[i][j] = 0.0; }
  qr[l][l] = 1.0;
  const double s2 = 0.7071067811865476;
  for (int m = 1; m <= l; ++m) {
    double sg = (m & 1) ? -1.0 : 1.0;
    qr[l + m][l + m] = sg * s2;
    qr[l + m][l - m] = s2;
    qi[l - m][l - m] = s2;
    qi[l - m][l + m] = -sg * s2;
  }
}

__global__ void cg_init_kernel(float* __restrict__ cg) {
  int t = threadIdx.x;
  if (t >= NCGT) return;
  const int L[NCGT][3] = {{0,0,0},{0,1,1},{0,2,2},{1,0,1},{1,1,0},{1,1,1},{1,1,2},
                          {1,2,1},{1,2,2},{2,0,2},{2,1,1},{2,1,2},{2,2,0},{2,2,1},{2,2,2}};
  int l1 = L[t][0], l2 = L[t][1], l3 = L[t][2];
  int n1 = 2 * l1 + 1, n2 = 2 * l2 + 1, n3 = 2 * l3 + 1;

  double C[5][5][5];
  for (int a = 0; a < 5; ++a)
    for (int b = 0; b < 5; ++b)
      for (int c = 0; c < 5; ++c) C[a][b][c] = 0.0;
  for (int m1 = -l1; m1 <= l1; ++m1)
    for (int m2 = -l2; m2 <= l2; ++m2) {
      int m3 = m1 + m2;
      if (m3 >= -l3 && m3 <= l3)
        C[l1 + m1][l2 + m2][l3 + m3] = cgc(l1, m1, l2, m2, l3, m3);
    }

  double q1r[5][5], q1i[5][5], q2r[5][5], q2i[5][5], q3r[5][5], q3i[5][5];
  qmat(l1, q1r, q1i); qmat(l2, q2r, q2i); qmat(l3, q3r, q3i);

  double Kr[5][5][5], Ki[5][5][5];
  double maxr = 0.0, maxi = 0.0;
  for (int i = 0; i < n1; ++i)
    for (int j = 0; j < n2; ++j)
      for (int k = 0; k < n3; ++k) {
        double sr = 0.0, si = 0.0;
        for (int a = 0; a < n1; ++a) {
          double ar = q1r[i][a], ai = -q1i[i][a];
          for (int b = 0; b < n2; ++b) {
            double br = q2r[j][b], bi = -q2i[j][b];
            double abr = ar * br - ai * bi;
            double abi = ar * bi + ai * br;
            for (int c = 0; c < n3; ++c) {
              double cc = C[a][b][c];
              if (cc == 0.0) continue;
              double cr = q3r[k][c], ci = q3i[k][c];
              sr += (abr * cr - abi * ci) * cc;
              si += (abr * ci + abi * cr) * cc;
            }
          }
        }
        Kr[i][j][k] = sr; Ki[i][j][k] = si;
        if (fabs(sr) > maxr) maxr = fabs(sr);
        if (fabs(si) > maxi) maxi = fabs(si);
      }
  bool useim = maxi > maxr;
  double norm = 0.0;
  for (int i = 0; i < n1; ++i)
    for (int j = 0; j < n2; ++j)
      for (int k = 0; k < n3; ++k) {
        double v = useim ? -Ki[i][j][k] : Kr[i][j][k];
        norm += v * v;
      }
  double scale = sqrt((double)n3 / norm);
  int off = cg_offset(l1, l2, l3);
  for (int i = 0; i < n1; ++i)
    for (int j = 0; j < n2; ++j)
      for (int k = 0; k < n3; ++k) {
        double v = useim ? -Ki[i][j][k] : Kr[i][j][k];
        cg[off + (i * n2 + j) * n3 + k] = (float)(v * scale);
      }
}

// ---------------------------------------------------------------------------
// Edge geometry: spherical harmonics (l=1: 3, l=2: 5) + radial embedding (10)
// ---------------------------------------------------------------------------
__global__ void geom_kernel(const float* __restrict__ edge_vec,
                            float* __restrict__ Yb, float* __restrict__ emb) {
  int e = blockIdx.x * blockDim.x + threadIdx.x;
  if (e >= NEDGES) return;
  float vx = edge_vec[e * 3 + 0];
  float vy = edge_vec[e * 3 + 1];
  float vz = edge_vec[e * 3 + 2];
  float r = sqrtf(vx * vx + vy * vy + vz * vz);
  float ir = 1.0f / r;
  float x = vx * ir, y = vy * ir, z = vz * ir;
  const float c3 = 1.7320508075688772f;
  const float c15 = 3.872983346207417f;
  const float c5 = 2.23606797749979f;
  float* Ye = Yb + (size_t)e * 8;
  Ye[0] = c3 * y;
  Ye[1] = c3 * z;
  Ye[2] = c3 * x;
  Ye[3] = c15 * x * y;
  Ye[4] = c15 * y * z;
  Ye[5] = 0.5f * c5 * (3.0f * z * z - 1.0f);
  Ye[6] = c15 * x * z;
  Ye[7] = 0.5f * c15 * (x * x - y * y);
  const float step = 5.0f / 11.0f;
  // 1.14136 * e^2 * sqrt(10)
  const float coef = 1.14136f * 7.38905609893065f * 3.1622776601683795f;
  for (int i = 0; i < 10; ++i) {
    float c = 5.0f * (float)(i + 1) / 11.0f;
    float d = (r - c) / step;
    float t1 = d + 1.0f, t2 = 1.0f - d;
    float u1 = (t1 > 0.0f) ? expf(-1.0f / t1) : 0.0f;
    float u2 = (t2 > 0.0f) ? expf(-1.0f / t2) : 0.0f;
    emb[(size_t)e * 10 + i] = coef * u1 * u2;
  }
}

// ---------------------------------------------------------------------------
// Radial MLP with fp32 WMMA:  wbuf[e,:] = silu(emb[e,:] @ W1) @ W2
//   emb: E x 10, W1: 10 x 100, W2: 100 x outw
// One block = 4 waves = one 16-edge tile. GEMM1 -> LDS hidden -> GEMM2.
// A(16x4 f32): lane M = lane%16, K = 4*ks + 2*half + vgpr
// C/D(16x16 f32): row = vgpr + 8*half, col = lane%16
// ---------------------------------------------------------------------------
__global__ __launch_bounds__(128) void radial_mlp_kernel(
    const float* __restrict__ emb, const float* __restrict__ w1,
    const float* __restrict__ w2, float* __restrict__ wbuf, int outw) {
  __shared__ float hid[16][112];
  const int wave = threadIdx.x >> 5;
  const int lane = threadIdx.x & 31;
  const int half = lane >> 4;
  const int lm = lane & 15;
  const size_t e0 = (size_t)blockIdx.x * 16;

  // Phase A: hidden = silu(emb_tile @ W1), 7 column tiles (112 padded cols)
  for (int nt = wave; nt < 7; nt += 4) {
    v8f c = {0.f, 0.f, 0.f, 0.f, 0.f, 0.f, 0.f, 0.f};
    for (int ks = 0; ks < 3; ++ks) {
      int k0 = ks * 4 + half * 2;
      v2f a, b;
      a.x = (k0 < 10) ? emb[(e0 + lm) * 10 + k0] : 0.0f;
      a.y = (k0 + 1 < 10) ? emb[(e0 + lm) * 10 + k0 + 1] : 0.0f;
      int ncol = nt * 16 + lm;
      b.x = (k0 < 10 && ncol < 100) ? w1[k0 * 100 + ncol] : 0.0f;
      b.y = (k0 + 1 < 10 && ncol < 100) ? w1[(k0 + 1) * 100 + ncol] : 0.0f;
      c = __builtin_amdgcn_wmma_f32_16x16x4_f32(false, a, false, b, (short)0, c,
                                                false, false);
    }
    int col = nt * 16 + lm;
    #pragma unroll
    for (int v = 0; v < 8; ++v) {
      int row = v + 8 * half;
      float xv = c[v];
      hid[row][col] = xv / (1.0f + expf(-xv));  // silu; padded cols -> silu(0)=0
    }
  }
  __syncthreads();

  // Phase B: out = hidden @ W2  (K = 100 exactly -> 25 WMMA steps)
  int ntiles = (outw + 15) >> 4;
  for (int nt = wave; nt < ntiles; nt += 4) {
    v8f c = {0.f, 0.f, 0.f, 0.f, 0.f, 0.f, 0.f, 0.f};
    int ncol = nt * 16 + lm;
    bool okc = (ncol < outw);
    for (int ks = 0; ks < 25; ++ks) {
      int k0 = ks * 4 + half * 2;
      v2f a, b;
      a.x = hid[lm][k0];
      a.y = hid[lm][k0 + 1];
      b.x = okc ? w2[k0 * outw + ncol] : 0.0f;
      b.y = okc ? w2[(k0 + 1) * outw + ncol] : 0.0f;
      c = __builtin_amdgcn_wmma_f32_16x16x4_f32(false, a, false, b, (short)0, c,
                                                false, false);
    }
    if (okc) {
      #pragma unroll
      for (int v = 0; v < 8; ++v) {
        int row = v + 8 * half;
        wbuf[(e0 + row) * (size_t)outw + ncol] = c[v];
      }
    }
  }
}

// ---------------------------------------------------------------------------
// Generic conv message + scatter (segment_sum via fp32 atomics)
// ---------------------------------------------------------------------------
struct PathDesc {
  int li, lf, lo, mul, w_off, u_off, cg_o;
};
struct ConvDesc {
  int npaths;
  int outw;
  PathDesc p[15];
};

__global__ void conv_msg_kernel(ConvDesc cd, const float* __restrict__ f0,
                                const float* __restrict__ f1,
                                const float* __restrict__ f2, int mul0, int mul1,
                                int mul2, const float* __restrict__ Yb,
                                const float* __restrict__ wbuf,
                                const float* __restrict__ cg,
                                const int* __restrict__ esrc,
                                const int* __restrict__ edst,
                                float* __restrict__ acc0, float* __restrict__ acc1,
                                float* __restrict__ acc2, int U0, int U1, int U2) {
  int e = blockIdx.x * blockDim.x + threadIdx.x;
  if (e >= NEDGES) return;
  int src = esrc[e];
  int dst = edst[e];
  float Y[9];
  Y[0] = 1.0f;
  {
    const float* Ye = Yb + (size_t)e * 8;
    for (int j = 0; j < 8; ++j) Y[1 + j] = Ye[j];
  }
  const float* we = wbuf + (size_t)e * cd.outw;
  for (int p = 0; p < cd.npaths; ++p) {
    const int li = cd.p[p].li, lf = cd.p[p].lf, lo = cd.p[p].lo;
    const int mul = cd.p[p].mul;
    const int ni = 2 * li + 1, nj = 2 * lf + 1, nk = 2 * lo + 1;
    const float* fbase;
    int fm;
    if (li == 0) { fbase = f0; fm = mul0; }
    else if (li == 1) { fbase = f1; fm = mul1; }
    else { fbase = f2; fm = mul2; }
    const float* fn = fbase + (size_t)src * fm * ni;
    const float* cgp = cg + cd.p[p].cg_o;
    const int yoff = (lf == 0) ? 0 : ((lf == 1) ? 1 : 4);
    float* accp;
    int Uw;
    if (lo == 0) { accp = acc0; Uw = U0; }
    else if (lo == 1) { accp = acc1; Uw = U1; }
    else { accp = acc2; Uw = U2; }
    for (int u = 0; u < mul; ++u) {
      float pw = we[cd.p[p].w_off + u];
      float fi[5];
      for (int i = 0; i < ni; ++i) fi[i] = fn[u * ni + i];
      float m[5] = {0.f, 0.f, 0.f, 0.f, 0.f};
      for (int i = 0; i < ni; ++i)
        for (int j = 0; j < nj; ++j) {
          float fy = fi[i] * Y[yoff + j];
          for (int k = 0; k < nk; ++k) m[k] += fy * cgp[(i * nj + j) * nk + k];
        }
      float* ap = accp + ((size_t)dst * Uw + cd.p[p].u_off + u) * nk;
      for (int k = 0; k < nk; ++k) atomicAdd(&ap[k], pw * m[k]);
    }
  }
}

// ---------------------------------------------------------------------------
// Node linear + optional self-connection: out[n,v,:] = inv*acc[n,:,:]@lin[:,v]
//                                                     + scf[n,:,:]@scw[:,v]
// One thread per (node, v).
// ---------------------------------------------------------------------------
__global__ void linsc_kernel(const float* __restrict__ acc, int U, int nk,
                             const float* __restrict__ lin, int V,
                             const float* __restrict__ scw,
                             const float* __restrict__ scf, int scU,
                             float* __restrict__ out, float inv) {
  int idx = blockIdx.x * blockDim.x + threadIdx.x;
  if (idx >= NNODES * V) return;
  int n = idx / V;
  int v = idx - n * V;
  float t[5] = {0.f, 0.f, 0.f, 0.f, 0.f};
  const float* ar = acc + (size_t)n * U * nk;
  for (int u = 0; u < U; ++u) {
    float w = lin[u * V + v];
    for (int k = 0; k < nk; ++k) t[k] += ar[u * nk + k] * w;
  }
  for (int k = 0; k < nk; ++k) t[k] *= inv;
  if (scw != nullptr) {
    const float* fr = scf + (size_t)n * scU * nk;
    for (int u = 0; u < scU; ++u) {
      float w = scw[u * V + v];
      for (int k = 0; k < nk; ++k) t[k] += fr[u * nk + k] * w;
    }
  }
  float* o = out + ((size_t)n * V + v) * nk;
  for (int k = 0; k < nk; ++k) o[k] = t[k];
}

// ---------------------------------------------------------------------------
// Gate: g = sigmoid(f0 @ gw); f1 *= g[:m1]; f2 *= g[m1:]; f0 = silu(f0)
// f0: [n][32], f1: [n][m1][3], f2: [n][m2][5], gw: 32 x (m1+m2)
// ---------------------------------------------------------------------------
__global__ void gate_kernel(float* __restrict__ f0, float* __restrict__ f1,
                            float* __restrict__ f2, const float* __restrict__ gw,
                            int m1, int m2) {
  int n = blockIdx.x * blockDim.x + threadIdx.x;
  if (n >= NNODES) return;
  const int m0 = 32;
  float s[32];
  for (int u = 0; u < m0; ++u) s[u] = f0[(size_t)n * m0 + u];
  int G = m1 + m2;
  for (int v = 0; v < m1; ++v) {
    float a = 0.f;
    for (int u = 0; u < m0; ++u) a += s[u] * gw[u * G + v];
    float g = 1.0f / (1.0f + expf(-a));
    float* r = f1 + ((size_t)n * m1 + v) * 3;
    for (int k = 0; k < 3; ++k) r[k] *= g;
  }
  for (int v = 0; v < m2; ++v) {
    float a = 0.f;
    for (int u = 0; u < m0; ++u) a += s[u] * gw[u * G + m1 + v];
    float g = 1.0f / (1.0f + expf(-a));
    float* r = f2 + ((size_t)n * m2 + v) * 5;
    for (int k = 0; k < 5; ++k) r[k] *= g;
  }
  for (int u = 0; u < m0; ++u)
    f0[(size_t)n * m0 + u] = s[u] / (1.0f + expf(-s[u]));
}

// ---------------------------------------------------------------------------
// Host side
// ---------------------------------------------------------------------------
static void build_desc(int layer, ConvDesc& cd, int U[3], const int muls[3]) {
  int plist[15][3];
  int np = 0;
  if (layer == 0) {
    int P[3][3] = {{0, 0, 0}, {0, 1, 1}, {0, 2, 2}};
    for (int i = 0; i < 3; ++i) {
      plist[np][0] = P[i][0]; plist[np][1] = P[i][1]; plist[np][2] = P[i][2]; ++np;
    }
  } else if (layer == 2) {
    int P[3][3] = {{0, 0, 0}, {1, 1, 0}, {2, 2, 0}};
    for (int i = 0; i < 3; ++i) {
      plist[np][0] = P[i][0]; plist[np][1] = P[i][1]; plist[np][2] = P[i][2]; ++np;
    }
  } else {
    for (int li = 0; li < 3; ++li)
      for (int lf = 0; lf < 3; ++lf)
        for (int lo = 0; lo < 3; ++lo) {
          int d = li - lf; if (d < 0) d = -d;
          if (d <= lo && lo <= li + lf) {
            plist[np][0] = li; plist[np][1] = lf; plist[np][2] = lo; ++np;
          }
        }
  }
  int woff = 0;
  int uoff[3] = {0, 0, 0};
  for (int i = 0; i < np; ++i) {
    int li = plist[i][0], lf = plist[i][1], lo = plist[i][2];
    int mul = muls[li];
    cd.p[i].li = li; cd.p[i].lf = lf; cd.p[i].lo = lo;
    cd.p[i].mul = mul; cd.p[i].w_off = woff; cd.p[i].u_off = uoff[lo];
    cd.p[i].cg_o = cg_offset(li, lf, lo);
    woff += mul;
    uoff[lo] += mul;
  }
  cd.npaths = np;
  cd.outw = woff;
  U[0] = uoff[0]; U[1] = uoff[1]; U[2] = uoff[2];
}

extern "C" void kernel_launch(void* const* d_in, const int* in_sizes, int n_in,
                              void* d_out, int out_size, void* d_ws, size_t ws_size,
                              hipStream_t stream) {
  const float* x = (const float*)d_in[0];
  const float* edge_vec = (const float*)d_in[1];
  const float* fc1_w1 = (const float*)d_in[2];
  const float* fc1_w2 = (const float*)d_in[3];
  const float* lin1_l0 = (const float*)d_in[4];
  const float* lin1_l1 = (const float*)d_in[5];
  const float* lin1_l2 = (const float*)d_in[6];
  const float* sc1_l0 = (const float*)d_in[7];
  const float* gate1_w = (const float*)d_in[8];
  const float* fc2_w1 = (const float*)d_in[9];
  const float* fc2_w2 = (const float*)d_in[10];
  const float* lin2_l0 = (const float*)d_in[11];
  const float* lin2_l1 = (const float*)d_in[12];
  const float* lin2_l2 = (const float*)d_in[13];
  const float* sc2_l0 = (const float*)d_in[14];
  const float* sc2_l1 = (const float*)d_in[15];
  const float* sc2_l2 = (const float*)d_in[16];
  const float* gate2_w = (const float*)d_in[17];
  const float* fc3_w1 = (const float*)d_in[18];
  const float* fc3_w2 = (const float*)d_in[19];
  const float* lin3_l0 = (const float*)d_in[20];
  const float* sc3_l0 = (const float*)d_in[21];
  const int* esrc = (const int*)d_in[22];
  const int* edst = (const int*)d_in[23];
  float* out = (float*)d_out;
  float* ws = (float*)d_ws;

  // workspace layout (floats)
  const size_t off_cg = 0;                                   // 615
  const size_t off_Y = 1024;                                 // E*8
  const size_t off_emb = off_Y + (size_t)NEDGES * 8;         // E*10
  const size_t off_w = off_emb + (size_t)NEDGES * 10;        // E*252
  const size_t off_acc0 = off_w + (size_t)NEDGES * 252;      // N*58
  const size_t off_acc1 = off_acc0 + (size_t)NNODES * 58;    // N*100*3
  const size_t off_acc2 = off_acc1 + (size_t)NNODES * 300;   // N*94*5
  const size_t off_h0 = off_acc2 + (size_t)NNODES * 470;     // N*32
  const size_t off_h1 = off_h0 + (size_t)NNODES * 32;        // N*16*3
  const size_t off_h2 = off_h1 + (size_t)NNODES * 48;        // N*10*5
  const size_t off_g0 = off_h2 + (size_t)NNODES * 50;
  const size_t off_g1 = off_g0 + (size_t)NNODES * 32;
  const size_t off_g2 = off_g1 + (size_t)NNODES * 48;

  float* cg = ws + off_cg;
  float* Yb = ws + off_Y;
  float* emb = ws + off_emb;
  float* wbuf = ws + off_w;
  float* acc0 = ws + off_acc0;
  float* acc1 = ws + off_acc1;
  float* acc2 = ws + off_acc2;
  float* h0 = ws + off_h0;
  float* h1 = ws + off_h1;
  float* h2 = ws + off_h2;
  float* g0 = ws + off_g0;
  float* g1 = ws + off_g1;
  float* g2 = ws + off_g2;

  const float inv = 0.25f;  // 1/sqrt(16)

  ConvDesc cd1, cd2, cd3;
  int U1a[3], U2a[3], U3a[3];
  const int muls_in[3] = {8, 0, 0};
  const int muls_hid[3] = {32, 16, 10};
  build_desc(0, cd1, U1a, muls_in);   // outw=24,  U={8,8,8}
  build_desc(1, cd2, U2a, muls_hid);  // outw=252, U={58,100,94}
  build_desc(2, cd3, U3a, muls_hid);  // outw=58,  U={58,0,0}

  const int EB = (NEDGES + 255) / 256;
  const int NB = (NNODES + 255) / 256;
  const int ETILES = NEDGES / 16;

  cg_init_kernel<<<1, 32, 0, stream>>>(cg);
  geom_kernel<<<EB, 256, 0, stream>>>(edge_vec, Yb, emb);

  // ---------------- Layer 1 ----------------
  radial_mlp_kernel<<<ETILES, 128, 0, stream>>>(emb, fc1_w1, fc1_w2, wbuf, cd1.outw);
  hipMemsetAsync(acc0, 0, (size_t)NNODES * 58 * sizeof(float), stream);
  hipMemsetAsync(acc1, 0, (size_t)NNODES * 300 * sizeof(float), stream);
  hipMemsetAsync(acc2, 0, (size_t)NNODES * 470 * sizeof(float), stream);
  conv_msg_kernel<<<(NEDGES + 127) / 128, 128, 0, stream>>>(
      cd1, x, x, x, 8, 0, 0, Yb, wbuf, cg, esrc, edst, acc0, acc1, acc2,
      U1a[0], U1a[1], U1a[2]);
  linsc_kernel<<<(NNODES * 32 + 255) / 256, 256, 0, stream>>>(
      acc0, U1a[0], 1, lin1_l0, 32, sc1_l0, x, 8, h0, inv);
  linsc_kernel<<<(NNODES * 16 + 255) / 256, 256, 0, stream>>>(
      acc1, U1a[1], 3, lin1_l1, 16, nullptr, nullptr, 0, h1, inv);
  linsc_kernel<<<(NNODES * 10 + 255) / 256, 256, 0, stream>>>(
      acc2, U1a[2], 5, lin1_l2, 10, nullptr, nullptr, 0, h2, inv);
  gate_kernel<<<NB, 256, 0, stream>>>(h0, h1, h2, gate1_w, 16, 10);

  // ---------------- Layer 2 ----------------
  radial_mlp_kernel<<<ETILES, 128, 0, stream>>>(emb, fc2_w1, fc2_w2, wbuf, cd2.outw);
  hipMemsetAsync(acc0, 0, (size_t)NNODES * 58 * sizeof(float), stream);
  hipMemsetAsync(acc1, 0, (size_t)NNODES * 300 * sizeof(float), stream);
  hipMemsetAsync(acc2, 0, (size_t)NNODES * 470 * sizeof(float), stream);
  conv_msg_kernel<<<(NEDGES + 127) / 128, 128, 0, stream>>>(
      cd2, h0, h1, h2, 32, 16, 10, Yb, wbuf, cg, esrc, edst, acc0, acc1, acc2,
      U2a[0], U2a[1], U2a[2]);
  linsc_kernel<<<(NNODES * 32 + 255) / 256, 256, 0, stream>>>(
      acc0, U2a[0], 1, lin2_l0, 32, sc2_l0, h0, 32, g0, inv);
  linsc_kernel<<<(NNODES * 16 + 255) / 256, 256, 0, stream>>>(
      acc1, U2a[1], 3, lin2_l1, 16, sc2_l1, h1, 16, g1, inv);
  linsc_kernel<<<(NNODES * 10 + 255) / 256, 256, 0, stream>>>(
      acc2, U2a[2], 5, lin2_l2, 10, sc2_l2, h2, 10, g2, inv);
  gate_kernel<<<NB, 256, 0, stream>>>(g0, g1, g2, gate2_w, 16, 10);

  // ---------------- Layer 3 ----------------
  radial_mlp_kernel<<<ETILES, 128, 0, stream>>>(emb, fc3_w1, fc3_w2, wbuf, cd3.outw);
  hipMemsetAsync(acc0, 0, (size_t)NNODES * 58 * sizeof(float), stream);
  conv_msg_kernel<<<(NEDGES + 127) / 128, 128, 0, stream>>>(
      cd3, g0, g1, g2, 32, 16, 10, Yb, wbuf, cg, esrc, edst, acc0, acc1, acc2,
      U3a[0], U3a[1], U3a[2]);
  linsc_kernel<<<(NNODES * 1 + 255) / 256, 256, 0, stream>>>(
      acc0, U3a[0], 1, lin3_l0, 1, sc3_l0, g0, 32, out, inv);
}